// basic_module_18786186953443
// MI455X (gfx1250) — compile-verified
//
#include <hip/hip_runtime.h>

#define BB   32
#define CC   512
#define BR   256
#define LL   4096
#define TILE 64
#define PSTR 72     // LDS row stride in floats (2*PSTR % 64 == 16 -> bank-split halves)
#define NPOS1 66    // stage-1 positions incl. +-1 halo
#define SBPAD 16    // extra floats so unguarded N-tile-4 reads stay in-bounds

typedef __attribute__((ext_vector_type(2))) float v2f;
typedef __attribute__((ext_vector_type(8))) float v8f;

__device__ __forceinline__ float mish_f(float x) {
  // mish(x) = x * tanh(softplus(x)) = x * ((1+e^x)^2 - 1) / ((1+e^x)^2 + 1)
  float t = __expf(x);
  float u = 1.f + t;
  float n = u * u - 1.f;
  float r = n / (n + 2.f);
  return (x > 15.f) ? x : x * r;
}

// K=256 fp32 GEMM accumulation for one wave: 2 M-tiles x NT N-tiles of 16x16,
// using V_WMMA_F32_16X16X4_F32 (A 16x4, B 4x16, C 16x16 fp32).
// A streamed from global weights (row-major 256x256), B read from LDS
// (row stride PSTR floats). No column guards: out-of-range B columns only
// affect D columns that are never stored (pad columns are zeroed).
template<int NT>
__device__ __forceinline__ void gemm_k256(const float* __restrict__ W,
                                          const float* sBl,
                                          int mb0, int mb1, int lane,
                                          v8f acc0[NT], v8f acc1[NT]) {
  const int c  = lane & 15;         // A: M row offset / B: N column
  const int kr = (lane >> 4) << 1;  // K sub-offset: 0 for lanes 0-15, 2 for 16-31
  const float* wp0 = W + (size_t)(mb0 + c) * BR + kr;
  const float* wp1 = W + (size_t)(mb1 + c) * BR + kr;
  const float* bp  = sBl + kr * PSTR + c;   // row kr, column c
#pragma unroll 2
  for (int k = 0; k < 64; ++k) {
    v2f a0 = *(const v2f*)(wp0 + 4 * k);  // (row, K=4k+kr), (row, K=4k+kr+1)
    v2f a1 = *(const v2f*)(wp1 + 4 * k);
    const float* bk = bp + (4 * k) * PSTR;
    v2f bf[NT];
#pragma unroll
    for (int n = 0; n < NT; ++n) {
      v2f t;
      t[0] = bk[n * 16];          // B[K=4k+kr  ][N=16n+c]
      t[1] = bk[n * 16 + PSTR];   // B[K=4k+kr+1][N=16n+c]
      bf[n] = t;
    }
#pragma unroll
    for (int n = 0; n < NT; ++n) {
      acc0[n] = __builtin_amdgcn_wmma_f32_16x16x4_f32(
          false, a0, false, bf[n], (short)0, acc0[n], false, false);
      acc1[n] = __builtin_amdgcn_wmma_f32_16x16x4_f32(
          false, a1, false, bf[n], (short)0, acc1[n], false, false);
    }
  }
}

__global__ __launch_bounds__(256)
void fused_shuffle_unit(const float* __restrict__ x,
                        const unsigned char* __restrict__ masks,
                        const float* __restrict__ w1,
                        const float* __restrict__ w2,
                        const float* __restrict__ w3,
                        const float* __restrict__ bn1w, const float* __restrict__ bn1b,
                        const float* __restrict__ bn1m, const float* __restrict__ bn1v,
                        const float* __restrict__ bn2w, const float* __restrict__ bn2b,
                        const float* __restrict__ bn2m, const float* __restrict__ bn2v,
                        const float* __restrict__ bn3w, const float* __restrict__ bn3b,
                        const float* __restrict__ bn3m, const float* __restrict__ bn3v,
                        float* __restrict__ out) {
  __shared__ float sB[BR * PSTR + SBPAD];  // x2 tile (A/B), then t2 (C/D)
  __shared__ float sT1[BR * PSTR];         // stage-1 result, 66 valid columns
  __shared__ float sc1[BR], sh1[BR], sc2[BR], sh2[BR], sc3[BR], sh3[BR];

  const int tid  = threadIdx.x;
  const int lane = tid & 31;
  const int wid  = tid >> 5;
  const int b  = blockIdx.x >> 6;          // 64 L-tiles per batch
  const int l0 = (blockIdx.x & 63) * TILE;

  // ---------------- Phase A: BN tables, x1 passthrough, x2 tile load -------
  {
    const int o = tid;  // blockDim == 256 == BR
    float s1 = bn1w[o] * rsqrtf(bn1v[o] + 1e-5f);
    sc1[o] = s1; sh1[o] = bn1b[o] - bn1m[o] * s1;
    float s2 = bn2w[o] * rsqrtf(bn2v[o] + 1e-5f);
    sc2[o] = s2; sh2[o] = bn2b[o] - bn2m[o] * s2;
    float s3 = bn3w[o] * rsqrtf(bn3v[o] + 1e-5f);
    sc3[o] = s3; sh3[o] = bn3b[o] - bn3m[o] * s3;
  }
  {
    // x1[k] -> output channel 2k+1 (channel-shuffle interleave)
    const int p = tid & 63;
    const int k0 = tid >> 6;
    const int l = l0 + p;
    for (int k = k0; k < BR; k += 4)
      out[((b * CC) + (2 * k + 1)) * LL + l] = x[((b * CC) + k) * LL + l];
  }
  for (int e = tid; e < BR * NPOS1; e += 256) {
    const int i = e / NPOS1;
    const int p = e - i * NPOS1;
    const int gl = l0 - 1 + p;
    float v = 0.f;
    if (gl >= 0 && gl < LL) v = x[((b * CC) + (BR + i)) * LL + gl];
    sB[i * PSTR + p] = v;
  }
  // zero pad columns 66..71 of every row + tail, so unguarded GEMM1 B reads
  // of the 5th N-tile see finite values (their D columns are discarded)
  for (int e = tid; e < BR * (PSTR - NPOS1) + SBPAD; e += 256) {
    if (e < BR * (PSTR - NPOS1)) {
      const int r = e / (PSTR - NPOS1);
      sB[r * PSTR + NPOS1 + (e - r * (PSTR - NPOS1))] = 0.f;
    } else {
      sB[BR * PSTR + (e - BR * (PSTR - NPOS1))] = 0.f;
    }
  }
  __syncthreads();

  const int mb0 = (2 * wid) * 16;       // wave's M-tile row bases
  const int mb1 = (2 * wid + 1) * 16;
  const int c   = lane & 15;
  const int hh  = lane >> 4;

  // ---------------- Phase B: GEMM1 + mish + BN1*mask -> sT1 ----------------
  {
    v8f acc0[5], acc1[5];
    const v8f vz = {0.f, 0.f, 0.f, 0.f, 0.f, 0.f, 0.f, 0.f};
#pragma unroll
    for (int n = 0; n < 5; ++n) { acc0[n] = vz; acc1[n] = vz; }
    gemm_k256<5>(w1, sB, mb0, mb1, lane, acc0, acc1);

    const unsigned char* mrow = masks + (size_t)b * LL;
#pragma unroll
    for (int mt = 0; mt < 2; ++mt) {
      const int rbase = (mt ? mb1 : mb0) + 8 * hh;
#pragma unroll
      for (int n = 0; n < 5; ++n) {
        const int p = n * 16 + c;
        if (p < NPOS1) {
          const int gl = l0 - 1 + p;
          const bool mk = (gl >= 0) && (gl < LL) && (mrow[gl] != 0);
          const v8f A = mt ? acc1[n] : acc0[n];
#pragma unroll
          for (int j = 0; j < 8; ++j) {
            const int row = rbase + j;
            const float m = mish_f(A[j]);
            sT1[row * PSTR + p] = mk ? (sc1[row] * m + sh1[row]) : 0.f;
          }
        }
      }
    }
  }
  __syncthreads();

  // ---------------- Phase C: depthwise conv(3) + BN2*mask -> sB (as t2) ----
  {
    const int p  = tid & 63;
    const int o0 = tid >> 6;
    const int l  = l0 + p;
    const bool mk = masks[(size_t)b * LL + l] != 0;
    for (int o = o0; o < BR; o += 4) {
      const float k0 = w2[o * 3 + 0];
      const float k1 = w2[o * 3 + 1];
      const float k2 = w2[o * 3 + 2];
      const float* t1r = sT1 + o * PSTR + p;
      const float s = k0 * t1r[0] + k1 * t1r[1] + k2 * t1r[2];
      sB[o * PSTR + p] = mk ? (sc2[o] * s + sh2[o]) : 0.f;
    }
  }
  __syncthreads();

  // ---------------- Phase D: GEMM2 + mish + BN3*mask -> even out channels --
  {
    v8f acc0[4], acc1[4];
    const v8f vz = {0.f, 0.f, 0.f, 0.f, 0.f, 0.f, 0.f, 0.f};
#pragma unroll
    for (int n = 0; n < 4; ++n) { acc0[n] = vz; acc1[n] = vz; }
    gemm_k256<4>(w3, sB, mb0, mb1, lane, acc0, acc1);

    const size_t obase = (size_t)b * CC * LL;
    const unsigned char* mrow = masks + (size_t)b * LL;
#pragma unroll
    for (int mt = 0; mt < 2; ++mt) {
      const int rbase = (mt ? mb1 : mb0) + 8 * hh;
#pragma unroll
      for (int n = 0; n < 4; ++n) {
        const int l = l0 + n * 16 + c;
        const bool mk = mrow[l] != 0;
        const v8f A = mt ? acc1[n] : acc0[n];
#pragma unroll
        for (int j = 0; j < 8; ++j) {
          const int row = rbase + j;
          const float m = mish_f(A[j]);
          out[obase + (size_t)(2 * row) * LL + l] =
              mk ? (sc3[row] * m + sh3[row]) : 0.f;
        }
      }
    }
  }
}

extern "C" void kernel_launch(void* const* d_in, const int* in_sizes, int n_in,
                              void* d_out, int out_size, void* d_ws, size_t ws_size,
                              hipStream_t stream) {
  const float* x  = (const float*)d_in[0];
  const unsigned char* masks = (const unsigned char*)d_in[1];  // numpy bool8
  const float* w1 = (const float*)d_in[2];
  const float* w2 = (const float*)d_in[3];
  const float* w3 = (const float*)d_in[4];
  const float* bn1w = (const float*)d_in[5];
  const float* bn1b = (const float*)d_in[6];
  const float* bn1m = (const float*)d_in[7];
  const float* bn1v = (const float*)d_in[8];
  const float* bn2w = (const float*)d_in[9];
  const float* bn2b = (const float*)d_in[10];
  const float* bn2m = (const float*)d_in[11];
  const float* bn2v = (const float*)d_in[12];
  const float* bn3w = (const float*)d_in[13];
  const float* bn3b = (const float*)d_in[14];
  const float* bn3m = (const float*)d_in[15];
  const float* bn3v = (const float*)d_in[16];
  float* out = (float*)d_out;

  dim3 grid(BB * (LL / TILE));  // 32 batches * 64 L-tiles = 2048 workgroups
  fused_shuffle_unit<<<grid, 256, 0, stream>>>(
      x, masks, w1, w2, w3,
      bn1w, bn1b, bn1m, bn1v,
      bn2w, bn2b, bn2m, bn2v,
      bn3w, bn3b, bn3m, bn3v,
      out);
}